// GNNModel_22110491640080
// MI455X (gfx1250) — compile-verified
//
#include <hip/hip_runtime.h>
#include <hip/hip_bf16.h>
#include <stdint.h>

// ---------------------------------------------------------------------------
// GCN on MI455X (gfx1250, wave32). fp32 WMMA (V_WMMA_F32_16X16X4_F32) for all
// dense GEMMs (32-row blocks, A-pair shares each B fragment); float4 gather +
// global_atomic_add_f32 scatter for edge aggregation; NT stores for the
// 614 MB of streaming head output. All hot indexing is 32-bit so the compiler
// can use SADDR + 32-bit voffset (GVS) addressing.
// ---------------------------------------------------------------------------

#define N_NODES   100000
#define N_EDGES   3200000
#define F_DIM     128
#define S_DIM     512

typedef __attribute__((ext_vector_type(2))) float v2f;
typedef __attribute__((ext_vector_type(8))) float v8f;

// ---------------- degree / normalization ----------------------------------
__global__ void k_init_deg(float* __restrict__ deg, float* __restrict__ gpart, int n) {
    uint i = blockIdx.x * blockDim.x + threadIdx.x;
    if (i < (uint)n) deg[i] = 1.0f;        // self loop
    if (i < F_DIM) gpart[i] = 0.0f;        // pool accumulator
}

__global__ void k_deg_accum(const long long* __restrict__ eidx, float* __restrict__ deg, int E) {
    uint e = blockIdx.x * blockDim.x + threadIdx.x;
    if (e >= (uint)E) return;
    int d = (int)eidx[(uint)E + e];        // edge_index[1] row
    atomicAdd(&deg[d], 1.0f);
}

__global__ void k_deg_rsqrt(float* __restrict__ deg, int n) {
    uint i = blockIdx.x * blockDim.x + threadIdx.x;
    if (i < (uint)n) deg[i] = rsqrtf(deg[i]);
}

// ---------------- WMMA GEMM: Y[N x Kout] = X[N x 128] @ W[128 x Kout] (+bias)
// Block = 256 threads (8 waves); covers 32 rows x Kout cols (Kout = 128*TPW).
// Two 16-row A fragments share every B fragment -> 2 WMMA per B load.
template <int TPW, bool NT>
__global__ __launch_bounds__(256) void k_gemm_xw(const float* __restrict__ X,
                                                 const float* __restrict__ W,
                                                 const float* __restrict__ bias,
                                                 float* __restrict__ Y,
                                                 int Kout) {
    __shared__ float lx[32 * 132];         // 32x128 X tile, padded stride 132
    const int tid  = threadIdx.x;
    const int row0 = blockIdx.x * 32;

    // cooperative load: 1024 float4 across 256 threads (32-bit offsets)
    const float4* X4 = (const float4*)X;
    for (int i = tid; i < 1024; i += 256) {
        int r  = i >> 5;                   // 32 float4 per row
        int c4 = i & 31;
        float4 v = X4[(uint)(row0 + r) * 32u + (uint)c4];
        float* p = &lx[r * 132 + c4 * 4];
        p[0] = v.x; p[1] = v.y; p[2] = v.z; p[3] = v.w;
    }
    __syncthreads();

    const int wave  = tid >> 5;
    const int lane  = tid & 31;
    const int nrow  = lane & 15;           // A: M (within tile) / B,D: N index
    const int khalf = lane >> 4;           // K pair select (A/B), M half (D)
    const int ncol0 = wave * (TPW * 16);

    v8f acc[2][TPW] = {};

    for (int k0 = 0; k0 < F_DIM; k0 += 4) {
        const int ka = k0 + khalf * 2;     // ISA 16x4 f32 A layout: K = 2*half + vgpr
        v2f a0, a1;
        a0.x = lx[nrow * 132 + ka];
        a0.y = lx[nrow * 132 + ka + 1];
        a1.x = lx[(nrow + 16) * 132 + ka];
        a1.y = lx[(nrow + 16) * 132 + ka + 1];
#pragma unroll
        for (int t = 0; t < TPW; t++) {
            const uint col = (uint)(ncol0 + t * 16 + nrow);
            v2f b;
            b.x = W[(uint)ka * (uint)Kout + col];
            b.y = W[(uint)(ka + 1) * (uint)Kout + col];
            acc[0][t] = __builtin_amdgcn_wmma_f32_16x16x4_f32(
                false, a0, false, b, (short)0, acc[0][t], false, false);
            acc[1][t] = __builtin_amdgcn_wmma_f32_16x16x4_f32(
                false, a1, false, b, (short)0, acc[1][t], false, false);
        }
    }

#pragma unroll
    for (int m = 0; m < 2; m++) {
#pragma unroll
        for (int t = 0; t < TPW; t++) {
            const uint col = (uint)(ncol0 + t * 16 + nrow);
            const float bb = bias ? bias[col] : 0.0f;
#pragma unroll
            for (int r = 0; r < 8; r++) {  // D layout: M = r + 8*khalf, N = nrow
                uint yo = (uint)(row0 + m * 16 + r + khalf * 8) * (uint)Kout + col;
                float v = acc[m][t][r] + bb;
                if (NT) __builtin_nontemporal_store(v, &Y[yo]);
                else    Y[yo] = v;
            }
        }
    }
}

// ---------------- aggregation ----------------------------------------------
// out[i,:] = xw[i,:] * dinv[i]^2   (self-loop term, initializes accumulator)
__global__ __launch_bounds__(256) void k_self_loop_init(const float* __restrict__ xw,
                                                        const float* __restrict__ dinv,
                                                        float* __restrict__ out) {
    uint idx = blockIdx.x * 256u + threadIdx.x;               // over N*32 float4
    if (idx >= (uint)N_NODES * 32u) return;
    uint row = idx >> 5;
    float s = dinv[row]; s = s * s;
    float4 v = ((const float4*)xw)[idx];
    float4 o; o.x = v.x * s; o.y = v.y * s; o.z = v.z * s; o.w = v.w * s;
    ((float4*)out)[idx] = o;
}

// one wave per edge: float4 gather, 4 f32 atomics per lane (128 feats/edge)
__global__ __launch_bounds__(256) void k_edge_scatter(const long long* __restrict__ eidx,
                                                      const float* __restrict__ dinv,
                                                      const float* __restrict__ xw,
                                                      float* __restrict__ out, int E) {
    uint e = (blockIdx.x * 256u + threadIdx.x) >> 5;
    if (e >= (uint)E) return;
    const uint lane = threadIdx.x & 31u;
    const int s = (int)eidx[e];
    const int d = (int)eidx[(uint)E + e];
    const float norm = dinv[s] * dinv[d];
    const float4 v = ((const float4*)xw)[(uint)s * 32u + lane];
    const uint oo = (uint)d * 128u + lane * 4u;
    atomicAdd(&out[oo + 0u], v.x * norm);
    atomicAdd(&out[oo + 1u], v.y * norm);
    atomicAdd(&out[oo + 2u], v.z * norm);
    atomicAdd(&out[oo + 3u], v.w * norm);
}

__global__ __launch_bounds__(256) void k_bias_relu(float* __restrict__ h,
                                                   const float* __restrict__ bias) {
    uint i = blockIdx.x * 256u + threadIdx.x;                 // over N*32 float4
    if (i >= (uint)N_NODES * 32u) return;
    float4 b = ((const float4*)bias)[i & 31u];
    float4 v = ((float4*)h)[i];
    v.x += b.x; v.y += b.y; v.z += b.z; v.w += b.w;
    v.x = v.x > 0.f ? v.x : 0.f;
    v.y = v.y > 0.f ? v.y : 0.f;
    v.z = v.z > 0.f ? v.z : 0.f;
    v.w = v.w > 0.f ? v.w : 0.f;
    ((float4*)h)[i] = v;
}

// ---------------- mean pool -------------------------------------------------
// block = 256 threads = 8 row-lanes x 32 float4-cols; 512 rows per block
__global__ __launch_bounds__(256) void k_pool_partial(const float* __restrict__ h,
                                                      float* __restrict__ gpart, int n) {
    __shared__ float4 red[256];
    const uint c4  = threadIdx.x & 31u;
    const uint sub = threadIdx.x >> 5;
    uint rbeg = blockIdx.x * 512u;
    uint rend = rbeg + 512u; if (rend > (uint)n) rend = (uint)n;
    const float4* h4 = (const float4*)h;
    float4 acc = {0.f, 0.f, 0.f, 0.f};
    for (uint r = rbeg + sub; r < rend; r += 8u) {
        float4 v = h4[r * 32u + c4];
        acc.x += v.x; acc.y += v.y; acc.z += v.z; acc.w += v.w;
    }
    red[threadIdx.x] = acc;
    __syncthreads();
    if (sub == 0) {
        float4 s = red[c4];
        for (int k = 1; k < 8; k++) {
            float4 v = red[k * 32 + c4];
            s.x += v.x; s.y += v.y; s.z += v.z; s.w += v.w;
        }
        atomicAdd(&gpart[c4 * 4u + 0u], s.x);
        atomicAdd(&gpart[c4 * 4u + 1u], s.y);
        atomicAdd(&gpart[c4 * 4u + 2u], s.z);
        atomicAdd(&gpart[c4 * 4u + 3u], s.w);
    }
}

// ---------------- tiny heads (g-based) --------------------------------------
__global__ void k_small_heads(const float* __restrict__ gpart,
                              const float* __restrict__ Wc,  const float* __restrict__ bc,
                              const float* __restrict__ Whl, const float* __restrict__ bhl,
                              const float* __restrict__ Wmt, const float* __restrict__ bmt,
                              const float* __restrict__ Wsl, const float* __restrict__ bsl,
                              float* __restrict__ out, size_t sl_off) {
    __shared__ float g[F_DIM];
    const int t = threadIdx.x;                 // 128 threads
    g[t] = gpart[t] * (1.0f / (float)N_NODES);
    __syncthreads();
    if (t == 0) {                                              // value [1]
        float s = 0.f;
        for (int f = 0; f < F_DIM; f++) s += g[f] * Wc[f];
        out[0] = s + bc[0];
    } else if (t >= 1 && t <= 4) {                             // high_level [4]
        int j = t - 1; float s = 0.f;
        for (int f = 0; f < F_DIM; f++) s += g[f] * Whl[f * 4 + j];
        out[1 + j] = s + bhl[j];
    } else if (t >= 5 && t <= 7) {                             // manage_line_type [3]
        int j = t - 5; float s = 0.f;
        for (int f = 0; f < F_DIM; f++) s += g[f] * Wmt[f * 3 + j];
        out[5 + j] = s + bmt[j];
    } else if (t >= 8 && t <= 15) {                            // select_line [8]
        int j = t - 8; float s = 0.f;
        for (int f = 0; f < F_DIM; f++) s += g[f] * Wsl[f * 8 + j];
        out[sl_off + j] = s + bsl[j];
    }
}

// ---------------------------------------------------------------------------
extern "C" void kernel_launch(void* const* d_in, const int* in_sizes, int n_in,
                              void* d_out, int out_size, void* d_ws, size_t ws_size,
                              hipStream_t stream) {
    const float*     X    = (const float*)d_in[0];
    const long long* eidx = (const long long*)d_in[1];
    const float *W1 = (const float*)d_in[2],  *b1  = (const float*)d_in[3];
    const float *W2 = (const float*)d_in[4],  *b2  = (const float*)d_in[5];
    const float *Wc = (const float*)d_in[6],  *bc  = (const float*)d_in[7];
    const float *Whl= (const float*)d_in[8],  *bhl = (const float*)d_in[9];
    const float *Wmt= (const float*)d_in[10], *bmt = (const float*)d_in[11];
    const float *Wp1= (const float*)d_in[12], *bp1 = (const float*)d_in[13];
    const float *Wp2= (const float*)d_in[14], *bp2 = (const float*)d_in[15];
    const float *Wdt= (const float*)d_in[16], *bdt = (const float*)d_in[17];
    const float *Wsl= (const float*)d_in[18], *bsl = (const float*)d_in[19];
    float* out = (float*)d_out;

    // workspace layout
    float* dinv  = (float*)d_ws;                           // N floats
    float* gpart = dinv + N_NODES;                         // 128 floats
    float* bufA  = gpart + F_DIM;                          // N*128 (xw)
    float* bufB  = bufA + (size_t)N_NODES * F_DIM;         // N*128 (h / agg)

    // output offsets (return-order concat)
    const size_t P1_OFF = 8;
    const size_t P2_OFF = P1_OFF + (size_t)N_NODES * S_DIM;
    const size_t DT_OFF = P2_OFF + (size_t)N_NODES * S_DIM;
    const size_t SL_OFF = DT_OFF + (size_t)N_NODES * S_DIM;

    const int gN   = (N_NODES + 255) / 256;
    const int gE   = (N_EDGES + 255) / 256;
    const int gEw  = (N_EDGES + 7) / 8;                    // 1 wave / edge
    const int gRow = N_NODES / 32;                         // 3125 (exact)
    const int gV4  = (N_NODES * 32 + 255) / 256;
    const int gP   = (N_NODES + 511) / 512;

    // degrees & symmetric normalization
    k_init_deg  <<<gN, 256, 0, stream>>>(dinv, gpart, N_NODES);
    k_deg_accum <<<gE, 256, 0, stream>>>(eidx, dinv, N_EDGES);
    k_deg_rsqrt <<<gN, 256, 0, stream>>>(dinv, N_NODES);

    // ---- layer 1: h1 = relu(agg(X@W1) + b1)
    k_gemm_xw<1, false> <<<gRow, 256, 0, stream>>>(X, W1, nullptr, bufA, F_DIM);
    k_self_loop_init    <<<gV4, 256, 0, stream>>>(bufA, dinv, bufB);
    k_edge_scatter      <<<gEw, 256, 0, stream>>>(eidx, dinv, bufA, bufB, N_EDGES);
    k_bias_relu         <<<gV4, 256, 0, stream>>>(bufB, b1);

    // ---- layer 2: h = relu(agg(h1@W2) + b2)
    k_gemm_xw<1, false> <<<gRow, 256, 0, stream>>>(bufB, W2, nullptr, bufA, F_DIM);
    k_self_loop_init    <<<gV4, 256, 0, stream>>>(bufA, dinv, bufB);
    k_edge_scatter      <<<gEw, 256, 0, stream>>>(eidx, dinv, bufA, bufB, N_EDGES);
    k_bias_relu         <<<gV4, 256, 0, stream>>>(bufB, b2);

    // ---- global mean pool + tiny heads
    k_pool_partial <<<gP, 256, 0, stream>>>(bufB, gpart, N_NODES);
    k_small_heads  <<<1, 128, 0, stream>>>(gpart, Wc, bc, Whl, bhl, Wmt, bmt,
                                           Wsl, bsl, out, SL_OFF);

    // ---- big per-node heads (N x 512 each), NT stores (write-once stream)
    k_gemm_xw<4, true> <<<gRow, 256, 0, stream>>>(bufB, Wp1, bp1, out + P1_OFF, S_DIM);
    k_gemm_xw<4, true> <<<gRow, 256, 0, stream>>>(bufB, Wp2, bp2, out + P2_OFF, S_DIM);
    k_gemm_xw<4, true> <<<gRow, 256, 0, stream>>>(bufB, Wdt, bdt, out + DT_OFF, S_DIM);
}